// BERTGraphModel_81793357185763
// MI455X (gfx1250) — compile-verified
//
#include <hip/hip_runtime.h>
#include <stdint.h>

// Problem constants (fixed by the harness / reference).
constexpr int NB = 32;     // B
constexpr int NT = 512;    // T
constexpr int ND = 1024;   // D
constexpr int NW = 100;    // W (word id == NW means "none")
constexpr int NS = NW + 1; // slots per batch: 1 CLS + NW words
constexpr int NL = 4;      // number of hidden layers

constexpr int CH = 3;                        // tokens per TDM chunk
constexpr int LDS_FLOATS = CH * NL * ND;     // 12288 floats = 48 KB staging buffer

typedef unsigned int u32x4 __attribute__((ext_vector_type(4)));
typedef int          i32x8 __attribute__((ext_vector_type(8)));
typedef int          i32x4 __attribute__((ext_vector_type(4)));

// One block (256 threads = 8 wave32) per output row (b * NS + s).
//  s == 0 : CLS = mean over 4 layers of hidden[:, t=0, :]
//  s >= 1 : word w = s-1 : mean over layers and over the contiguous token
//           range whose sorted word_ids equal w.
__global__ __launch_bounds__(256) void
bert_seg_mean_tdm(const float* __restrict__ hidden,   // (NL, NB, NT, ND)
                  const int*   __restrict__ word_ids, // (NB, NT), sorted rows
                  float*       __restrict__ out)      // (NB*NS, ND)
{
    __shared__ __align__(16) float buf[LDS_FLOATS];

    const int row = blockIdx.x;          // 0 .. NB*NS-1
    const int b   = row / NS;
    const int s   = row - b * NS;
    const int tid = threadIdx.x;         // 0..255, owns d = 4*tid..4*tid+3

    // ---- token range [t0, t1) for this output row (ids sorted per batch) ----
    int t0, t1;
    if (s == 0) {
        t0 = 0; t1 = 1;                  // CLS: token 0 only
    } else {
        const int w = s - 1;
        const int* ids = word_ids + b * NT;
        int lo = 0, hi = NT;             // lower_bound(w)
        while (lo < hi) { int m = (lo + hi) >> 1; if (ids[m] < w) lo = m + 1; else hi = m; }
        t0 = lo;
        hi = NT;                         // lower_bound(w+1)
        while (lo < hi) { int m = (lo + hi) >> 1; if (ids[m] < w + 1) lo = m + 1; else hi = m; }
        t1 = lo;
    }
    const int count = t1 - t0;

    const uint32_t lds_addr = (uint32_t)(uintptr_t)(&buf[0]);

    float ax = 0.f, ay = 0.f, az = 0.f, aw = 0.f;

    for (int c0 = 0; c0 < count; c0 += CH) {
        const int ch = (count - c0 < CH) ? (count - c0) : CH;

        if (tid < 32) {  // wave 0 drives the TDM (EXEC is ignored by tensor ops)
            // Tile start: hidden[layer=0][b][t0+c0][0]; 3-D tile:
            //   X (dim0) = ND elements, Y (dim1) = ch tokens (stride ND),
            //   Z (dim2) = NL layers   (stride NB*NT*ND).
            const uint64_t gaddr =
                (uint64_t)(uintptr_t)(hidden + ((size_t)b * NT + (size_t)(t0 + c0)) * ND);

            u32x4 g0;
            g0[0] = 1u;                                   // count=1 valid D#, load, user mode
            g0[1] = lds_addr;                             // LDS byte address
            g0[2] = (uint32_t)gaddr;                      // global_addr[31:0]
            g0[3] = (uint32_t)((gaddr >> 32) & 0x1FFFFFFu) | 0x80000000u; // [56:32] | type=2

            i32x8 g1;
            g1[0] = 0x00020000;                           // data_size = 4 bytes (code 2)
            g1[1] = (int)((uint32_t)ND << 16);            // tensor_dim0 lo16 (=1024) at [63:48]
            g1[2] = (int)(16384u << 16);                  // dim0 hi16=0 | tensor_dim1 lo16=16384
            g1[3] = (int)((uint32_t)ND << 16);            // dim1 hi16=0 | tile_dim0 = 1024
            g1[4] = ch | (NL << 16);                      // tile_dim1 = ch | tile_dim2 = NL
            g1[5] = ND;                                   // tensor_dim0_stride lo32 = 1024
            g1[6] = 0;                                    // dim0_stride hi16 | dim1_stride lo16 (16M -> 0)
            g1[7] = (int)(((uint64_t)NB * NT * ND) >> 16);// dim1_stride [47:16] = 256

            i32x4 g2;
            g2[0] = NL;                                   // tensor_dim2 = 4
            g2[1] = 0;                                    // tensor_dim3 = 0 (unused)
            g2[2] = 0;                                    // dim2_stride lo32
            g2[3] = 0;                                    // dim2_stride hi | tile_dim3 = 0

            i32x4 g3;
            g3[0] = 0; g3[1] = 0; g3[2] = 0; g3[3] = 0;   // dims 3/4 unused

            i32x8 g4;                                     // VADDR4 slot: unused, zero
            g4[0] = 0; g4[1] = 0; g4[2] = 0; g4[3] = 0;
            g4[4] = 0; g4[5] = 0; g4[6] = 0; g4[7] = 0;

            __builtin_amdgcn_tensor_load_to_lds(g0, g1, g2, g3, g4, /*cpol=*/0);
            __builtin_amdgcn_s_wait_tensorcnt(0);         // DMA landed in LDS
        }
        __syncthreads();                                   // publish staged rows to all waves

        const int R = ch * NL;                             // 4 KB rows staged this chunk
        #pragma unroll 4
        for (int r = 0; r < R; ++r) {
            const float4 v = *reinterpret_cast<const float4*>(&buf[r * ND + (tid << 2)]);
            ax += v.x; ay += v.y; az += v.z; aw += v.w;
        }
        __syncthreads();                                   // done reading before next DMA
    }

    // ---- normalize and store (count==0 -> zeros, matching the reference) ----
    const float inv = 1.0f / ((float)NL * (float)(count > 0 ? count : 1));
    float4 r4;
    r4.x = ax * inv; r4.y = ay * inv; r4.z = az * inv; r4.w = aw * inv;
    *reinterpret_cast<float4*>(out + (size_t)row * ND + (size_t)(tid << 2)) = r4;
}

extern "C" void kernel_launch(void* const* d_in, const int* in_sizes, int n_in,
                              void* d_out, int out_size, void* d_ws, size_t ws_size,
                              hipStream_t stream) {
    (void)in_sizes; (void)n_in; (void)out_size; (void)d_ws; (void)ws_size;
    const float* hidden   = (const float*)d_in[0];
    const int*   word_ids = (const int*)d_in[1];
    // d_in[2] = num_words scalar (== NW), baked in as a compile-time constant.
    float* out = (float*)d_out;

    bert_seg_mean_tdm<<<dim3(NB * NS), dim3(256), 0, stream>>>(hidden, word_ids, out);
}